// CNN_LeNet_83794811945244
// MI455X (gfx1250) — compile-verified
//
#include <hip/hip_runtime.h>
#include <hip/hip_bf16.h>

typedef __attribute__((ext_vector_type(2))) float v2f;
typedef __attribute__((ext_vector_type(8))) float v8f;

#define BN_EPS 1e-5f

// ---------------------------------------------------------------- init stats
__global__ void k_zero_stats(float* __restrict__ s) {
    if (threadIdx.x < 64) s[threadIdx.x] = 0.0f;
}

// ---------------------------------------------------------------- adder layer 1
// x:[1024,1,32,32]  w1:[6,1,5,5]  h1:[1024,6,28,28]=relu(-sum|x-w|)
// stats[0..5]=sum per channel, stats[8..13]=sumsq per channel (post-ReLU)
__global__ __launch_bounds__(256) void k_adder1(const float* __restrict__ x,
                                                const float* __restrict__ w1,
                                                float* __restrict__ h1,
                                                float* __restrict__ stats) {
    __shared__ float xs[1024];
    __shared__ float wsh[150];
    __shared__ float ssum[6], sssq[6];
    const int b = blockIdx.x;
    const int tid = threadIdx.x;
    for (int i = tid; i < 1024; i += 256) xs[i] = x[b * 1024 + i];
    if (tid < 150) wsh[tid] = w1[tid];
    if (tid < 6) { ssum[tid] = 0.f; sssq[tid] = 0.f; }
    __syncthreads();

    float ls[6] = {0.f,0.f,0.f,0.f,0.f,0.f};
    float lq[6] = {0.f,0.f,0.f,0.f,0.f,0.f};
    for (int pos = tid; pos < 784; pos += 256) {
        const int p = pos / 28, q = pos % 28;
        float acc[6] = {0.f,0.f,0.f,0.f,0.f,0.f};
        #pragma unroll
        for (int kh = 0; kh < 5; ++kh) {
            #pragma unroll
            for (int kw = 0; kw < 5; ++kw) {
                const float xv = xs[(p + kh) * 32 + (q + kw)];
                const int t = kh * 5 + kw;
                #pragma unroll
                for (int o = 0; o < 6; ++o)
                    acc[o] += fabsf(xv - wsh[o * 25 + t]);
            }
        }
        #pragma unroll
        for (int o = 0; o < 6; ++o) {
            const float r = fmaxf(0.0f, -acc[o]);
            h1[((b * 6 + o) * 28 + p) * 28 + q] = r;
            ls[o] += r; lq[o] += r * r;
        }
    }
    #pragma unroll
    for (int o = 0; o < 6; ++o) {
        atomicAdd(&ssum[o], ls[o]);
        atomicAdd(&sssq[o], lq[o]);
    }
    __syncthreads();
    if (tid < 6) {
        atomicAdd(&stats[tid], ssum[tid]);
        atomicAdd(&stats[8 + tid], sssq[tid]);
    }
}

// ---------------------------------------------------------------- BN finalize
__global__ void k_bn_finalize(const float* __restrict__ ssum,
                              const float* __restrict__ sssq,
                              const float* __restrict__ gamma,
                              const float* __restrict__ beta,
                              float* __restrict__ scale,
                              float* __restrict__ shift,
                              int C, float invN) {
    const int c = threadIdx.x;
    if (c < C) {
        const float mean = ssum[c] * invN;
        const float var  = fmaxf(sssq[c] * invN - mean * mean, 0.0f);
        const float sc   = gamma[c] * rsqrtf(var + BN_EPS);
        scale[c] = sc;
        shift[c] = beta[c] - mean * sc;
    }
}

// ---------------------------------------------------------------- BN affine + 2x2 maxpool (28->14)
__global__ void k_pool1(const float* __restrict__ h1, const float* __restrict__ scale,
                        const float* __restrict__ shift, float* __restrict__ p1, int total) {
    const int idx = blockIdx.x * blockDim.x + threadIdx.x;
    if (idx >= total) return;
    const int q = idx % 14;
    const int p = (idx / 14) % 14;
    const int o = (idx / 196) % 6;
    const int b = idx / 1176;
    const float sc = scale[o], sh = shift[o];
    const float* base = h1 + ((b * 6 + o) * 28 + 2 * p) * 28 + 2 * q;
    const float v0 = sc * base[0]  + sh;
    const float v1 = sc * base[1]  + sh;
    const float v2 = sc * base[28] + sh;
    const float v3 = sc * base[29] + sh;
    p1[idx] = fmaxf(fmaxf(v0, v1), fmaxf(v2, v3));
}

// ---------------------------------------------------------------- adder layer 2
// p1:[1024,6,14,14]  w2:[16,6,5,5]  h2:[1024,16,10,10]
// stats[16..31]=sum, stats[32..47]=sumsq
__global__ __launch_bounds__(256) void k_adder2(const float* __restrict__ p1,
                                                const float* __restrict__ w2,
                                                float* __restrict__ h2,
                                                float* __restrict__ stats) {
    __shared__ float xs[1176];
    __shared__ float wsh[2400];
    __shared__ float ssum[16], sssq[16];
    const int b = blockIdx.x;
    const int tid = threadIdx.x;
    for (int i = tid; i < 1176; i += 256) xs[i] = p1[b * 1176 + i];
    for (int i = tid; i < 2400; i += 256) wsh[i] = w2[i];
    if (tid < 16) { ssum[tid] = 0.f; sssq[tid] = 0.f; }
    __syncthreads();

    for (int item = tid; item < 1600; item += 256) {
        const int o = item / 100;
        const int pos = item % 100;
        const int p = pos / 10, q = pos % 10;
        float acc = 0.f;
        for (int c = 0; c < 6; ++c) {
            const float* xc = &xs[c * 196 + p * 14 + q];
            const float* wc = &wsh[o * 150 + c * 25];
            #pragma unroll
            for (int kh = 0; kh < 5; ++kh) {
                #pragma unroll
                for (int kw = 0; kw < 5; ++kw)
                    acc += fabsf(xc[kh * 14 + kw] - wc[kh * 5 + kw]);
            }
        }
        const float r = fmaxf(0.0f, -acc);
        h2[b * 1600 + item] = r;     // item == o*100 + p*10 + q, matches [o][10][10]
        atomicAdd(&ssum[o], r);
        atomicAdd(&sssq[o], r * r);
    }
    __syncthreads();
    if (tid < 16) {
        atomicAdd(&stats[16 + tid], ssum[tid]);
        atomicAdd(&stats[32 + tid], sssq[tid]);
    }
}

// ---------------------------------------------------------------- BN affine + 2x2 maxpool (10->5), flatten [o][5][5]
__global__ void k_pool2(const float* __restrict__ h2, const float* __restrict__ scale,
                        const float* __restrict__ shift, float* __restrict__ p2, int total) {
    const int idx = blockIdx.x * blockDim.x + threadIdx.x;
    if (idx >= total) return;
    const int q = idx % 5;
    const int p = (idx / 5) % 5;
    const int o = (idx / 25) % 16;
    const int b = idx / 400;
    const float sc = scale[o], sh = shift[o];
    const float* base = h2 + b * 1600 + o * 100 + (2 * p) * 10 + 2 * q;
    const float v0 = sc * base[0]  + sh;
    const float v1 = sc * base[1]  + sh;
    const float v2 = sc * base[10] + sh;
    const float v3 = sc * base[11] + sh;
    p2[idx] = fmaxf(fmaxf(v0, v1), fmaxf(v2, v3));
}

// ---------------------------------------------------------------- FC via f32 WMMA 16x16x4
// Y[M=1024, Nact] = act(X[M,K] @ W[Nact,K]^T + bias), one wave per 16x16 tile.
// A lane layout (16x4 f32): lane<16 -> M=lane, K={0,1}; lane>=16 -> M=lane-16, K={2,3}
// B lane layout (4x16):     lane<16 -> N=lane, K={0,1}; lane>=16 -> N=lane-16, K={2,3}
// D layout: VGPR r -> row r + 8*(lane>=16), col = lane&15
//
// N-padding: lanes with n >= Nact load a *clamped* valid weight row instead of
// branching — their D columns are garbage but are never stored. This keeps the
// K-loop branch-free (EXEC all-1s, no saveexec per load) and lets loads pipeline.
__global__ __launch_bounds__(32) void k_fc_wmma(const float* __restrict__ X,
                                                const float* __restrict__ W,
                                                const float* __restrict__ bias,
                                                float* __restrict__ Y,
                                                int K, int Nact, int ntN, int do_relu) {
    const int tile = blockIdx.x;
    const int tm = tile / ntN;
    const int tn = tile % ntN;
    const int lane = threadIdx.x;
    const int half = lane >> 4;
    const int l16 = lane & 15;
    const int mrow = tm * 16 + l16;
    const int nrow = tn * 16 + l16;
    const int nclamp = nrow < Nact ? nrow : (Nact - 1);   // in-bounds row for all lanes
    const int koff = half * 2;
    const float* arow = X + (size_t)mrow * K + koff;
    const float* brow = W + (size_t)nclamp * K + koff;

    v8f acc = {0.f, 0.f, 0.f, 0.f, 0.f, 0.f, 0.f, 0.f};
    for (int k = 0; k < K; k += 4) {
        v2f a  = *(const v2f*)(arow + k);
        v2f bm = *(const v2f*)(brow + k);
        acc = __builtin_amdgcn_wmma_f32_16x16x4_f32(
            /*neg_a=*/false, a, /*neg_b=*/false, bm,
            /*c_mod=*/(short)0, acc, /*reuse_a=*/false, /*reuse_b=*/false);
    }

    if (nrow < Nact) {
        const float bv = bias[nrow];
        #pragma unroll
        for (int r = 0; r < 8; ++r) {
            const int m = tm * 16 + r + 8 * half;
            float y = acc[r] + bv;
            if (do_relu) y = fmaxf(y, 0.f);
            Y[(size_t)m * Nact + nrow] = y;
        }
    }
}

// ---------------------------------------------------------------- softmax over 10 classes
__global__ void k_softmax(const float* __restrict__ z, float* __restrict__ out, int rows) {
    const int r = blockIdx.x * blockDim.x + threadIdx.x;
    if (r >= rows) return;
    const float* zr = z + r * 10;
    float mx = zr[0];
    #pragma unroll
    for (int j = 1; j < 10; ++j) mx = fmaxf(mx, zr[j]);
    float e[10];
    float s = 0.f;
    #pragma unroll
    for (int j = 0; j < 10; ++j) { e[j] = expf(zr[j] - mx); s += e[j]; }
    const float inv = 1.0f / s;
    #pragma unroll
    for (int j = 0; j < 10; ++j) out[r * 10 + j] = e[j] * inv;
}

// ---------------------------------------------------------------- launch
extern "C" void kernel_launch(void* const* d_in, const int* in_sizes, int n_in,
                              void* d_out, int out_size, void* d_ws, size_t ws_size,
                              hipStream_t stream) {
    const float* x    = (const float*)d_in[0];
    const float* w1   = (const float*)d_in[1];
    const float* g1   = (const float*)d_in[2];
    const float* b1   = (const float*)d_in[3];
    const float* w2   = (const float*)d_in[4];
    const float* g2   = (const float*)d_in[5];
    const float* b2   = (const float*)d_in[6];
    const float* fc1w = (const float*)d_in[7];
    const float* fc1b = (const float*)d_in[8];
    const float* fc2w = (const float*)d_in[9];
    const float* fc2b = (const float*)d_in[10];
    const float* fc3w = (const float*)d_in[11];
    const float* fc3b = (const float*)d_in[12];
    float* out = (float*)d_out;

    float* ws     = (float*)d_ws;
    float* stats  = ws;          // [0..7] bn1 sum, [8..15] bn1 ssq, [16..31] bn2 sum, [32..47] bn2 ssq
    float* affine = ws + 64;     // [0..7] bn1 scale, [8..15] bn1 shift, [16..31] bn2 scale, [32..47] bn2 shift
    float* h1 = ws + 128;                          // 1024*6*28*28
    float* p1 = h1 + 1024 * 6 * 28 * 28;           // 1024*6*14*14
    float* h2 = p1 + 1024 * 6 * 14 * 14;           // 1024*16*10*10
    float* p2 = h2 + 1024 * 16 * 10 * 10;          // 1024*400
    float* a1 = p2 + 1024 * 400;                   // 1024*120
    float* a2 = a1 + 1024 * 120;                   // 1024*84
    float* z3 = a2 + 1024 * 84;                    // 1024*10

    k_zero_stats<<<1, 64, 0, stream>>>(stats);

    k_adder1<<<1024, 256, 0, stream>>>(x, w1, h1, stats);
    k_bn_finalize<<<1, 32, 0, stream>>>(stats, stats + 8, g1, b1,
                                        affine, affine + 8, 6,
                                        1.0f / (1024.0f * 784.0f));
    {
        const int total = 1024 * 6 * 14 * 14;
        k_pool1<<<(total + 255) / 256, 256, 0, stream>>>(h1, affine, affine + 8, p1, total);
    }

    k_adder2<<<1024, 256, 0, stream>>>(p1, w2, h2, stats);
    k_bn_finalize<<<1, 32, 0, stream>>>(stats + 16, stats + 32, g2, b2,
                                        affine + 16, affine + 32, 16,
                                        1.0f / (1024.0f * 100.0f));
    {
        const int total = 1024 * 400;
        k_pool2<<<(total + 255) / 256, 256, 0, stream>>>(h2, affine + 16, affine + 32, p2, total);
    }

    k_fc_wmma<<<64 * 8, 32, 0, stream>>>(p2, fc1w, fc1b, a1, 400, 120, 8, 1);
    k_fc_wmma<<<64 * 6, 32, 0, stream>>>(a1, fc2w, fc2b, a2, 120,  84, 6, 1);
    k_fc_wmma<<<64 * 1, 32, 0, stream>>>(a2, fc3w, fc3b, z3,  84,  10, 1, 0);

    k_softmax<<<(1024 + 255) / 256, 256, 0, stream>>>(z3, out, 1024);
}